// PointCloudRenderer_6064493822058
// MI455X (gfx1250) — compile-verified
//
#include <hip/hip_runtime.h>
#include <hip/hip_bf16.h>

#define IMGW 128
#define PN   2048
#define BN   2
#define KN   10
#define RADN 0.005f
#define BIGZ 1e10f

typedef float v2f __attribute__((ext_vector_type(2)));
typedef float v8f __attribute__((ext_vector_type(8)));
typedef unsigned int u32x4 __attribute__((ext_vector_type(4)));
typedef int i32x8 __attribute__((ext_vector_type(8)));
typedef int i32x4 __attribute__((ext_vector_type(4)));

#if __has_builtin(__builtin_amdgcn_tensor_load_to_lds)
#define HAS_TDM 1
#else
#define HAS_TDM 0
#endif

// ---------------------------------------------------------------------------
// Kernel 1: project points -> record {-2U, -2V, U^2+V^2, z} plus contiguous z[]
// (U,V are pixel coords scaled by 2/IMG so dist^2 is directly in NDC units.)
// ---------------------------------------------------------------------------
__global__ __launch_bounds__(256) void pcr_project(
    const float* __restrict__ pts, const float* __restrict__ R,
    const float* __restrict__ t, const float* __restrict__ fptr,
    float4* __restrict__ rec, float* __restrict__ zarr) {
  int i = blockIdx.x * blockDim.x + threadIdx.x;
  if (i >= BN * PN) return;
  int b = i / PN;
  const float* p = pts + (size_t)i * 3;
  float x = p[0], y = p[1], w = p[2];
  const float* Rb = R + b * 9;
  const float* tb = t + b * 3;
  // x_cam = x_world @ R + t  (row-vector convention): cam_j = sum_i p_i R[i][j]
  float cx = x * Rb[0] + y * Rb[3] + w * Rb[6] + tb[0];
  float cy = x * Rb[1] + y * Rb[4] + w * Rb[7] + tb[1];
  float cz = x * Rb[2] + y * Rb[5] + w * Rb[8] + tb[2];
  float f = fptr[0];
  float inv = 1.0f / cz;
  const float scale = 2.0f / (float)IMGW;
  float U = (f * cx * inv + 0.5f * IMGW) * scale;
  float V = (f * cy * inv + 0.5f * IMGW) * scale;
  rec[i] = make_float4(-2.0f * U, -2.0f * V, U * U + V * V, cz);
  zarr[i] = cz;
}

// ---------------------------------------------------------------------------
// TDM: 1-D dword tile, global -> LDS.  D# built per CDNA5 ISA sec 8.
// 6-arg builtin form (clang-23 / therock-10.0): (g0, g1, g2, g3, g4, cpol).
// ---------------------------------------------------------------------------
__device__ __forceinline__ void tdm_load_1d(unsigned lds_addr, const void* gptr,
                                            unsigned n_dw) {
#if HAS_TDM
  unsigned long long ga = (unsigned long long)(uintptr_t)gptr;
  u32x4 g0;
  g0[0] = 1u;                                              // count=1, user mode
  g0[1] = lds_addr;                                        // LDS byte address
  g0[2] = (unsigned)ga;                                    // global_addr[31:0]
  g0[3] = ((unsigned)(ga >> 32) & 0x01FFFFFFu) | 0x80000000u; // [56:32] | type=2
  i32x8 g1;
  g1[0] = (int)(2u << 16);                                 // data_size = 4B
  g1[1] = (int)((n_dw & 0xFFFFu) << 16);                   // tensor_dim0 lo
  g1[2] = (int)((n_dw >> 16) | (1u << 16));                // tensor_dim0 hi, dim1=1
  g1[3] = (int)((n_dw & 0xFFFFu) << 16);                   // tile_dim0 = n_dw
  g1[4] = 0;                                               // tile_dim1/2 unused
  g1[5] = (int)n_dw;                                       // tensor_dim0_stride lo
  g1[6] = 0;
  g1[7] = 0;
  i32x4 gz4 = {0, 0, 0, 0};
  i32x8 gz8 = {0, 0, 0, 0, 0, 0, 0, 0};
  __builtin_amdgcn_tensor_load_to_lds(g0, g1, gz4, gz4, gz8, 0);
#else
  (void)lds_addr; (void)gptr; (void)n_dw;
#endif
}

// Branchless sorted-ascending insertion into the per-lane K-list.
// Ternaries lower to v_cndmask (no exec-mask save/restore, no array copies).
#define INSERT_K(zv, dv, iv)                                   \
  {                                                            \
    float _cz = (zv), _cd = (dv); int _ci = (iv);              \
    _Pragma("unroll")                                          \
    for (int _s = 0; _s < KN; ++_s) {                          \
      bool _lt = _cz < zk[_s];                                 \
      float _tz = zk[_s]; zk[_s] = _lt ? _cz : _tz; _cz = _lt ? _tz : _cz; \
      float _td = d2k[_s]; d2k[_s] = _lt ? _cd : _td; _cd = _lt ? _td : _cd; \
      int _ti = ik[_s]; ik[_s] = _lt ? _ci : _ti; _ci = _lt ? _ti : _ci; \
    }                                                          \
  }

// ---------------------------------------------------------------------------
// Kernel 2: one wave per 16 pixels; WMMA 16x16x4 computes dist^2 for a
// 16-pixel x 16-point tile; per-lane z-buffer insert; shfl-merge; composite.
// ---------------------------------------------------------------------------
__global__ __launch_bounds__(256) void pcr_render(
    const float4* __restrict__ rec, const float* __restrict__ zarr,
    const float* __restrict__ feat, float* __restrict__ out) {
  __shared__ float4 sRec[PN];       // 32 KB: {-2U,-2V,q,z} per point
  __shared__ float4 sZ4[PN / 4];    //  8 KB: z packed as float4

  const int b = blockIdx.y;
  const int tid = threadIdx.x;
  const int wv = tid >> 5;
  const int lane = tid & 31;
  const bool hi = (lane & 16) != 0; // lane half: K2/K3 rows & points 8-15
  const int l16 = lane & 15;

#if HAS_TDM
  if (tid < 32) {
    tdm_load_1d((unsigned)(uintptr_t)&sRec[0], rec + (size_t)b * PN, PN * 4);
    tdm_load_1d((unsigned)(uintptr_t)&sZ4[0], zarr + (size_t)b * PN, PN);
#if __has_builtin(__builtin_amdgcn_s_wait_tensorcnt)
    __builtin_amdgcn_s_wait_tensorcnt(0);
#endif
  }
#else
  for (int i = tid; i < PN; i += 256) {
    sRec[i] = rec[(size_t)b * PN + i];
    ((float*)sZ4)[i] = zarr[(size_t)b * PN + i];
  }
#endif
  __syncthreads();

  const float scale = 2.0f / (float)IMGW;
  const int pix = (blockIdx.x * 8 + wv) * 16 + l16;   // this lane's pixel column
  const int ix = pix & (IMGW - 1), iy = pix >> 7;     // n = iy*IMG + ix (ij meshgrid)
  const float Ug = ((float)ix + 0.5f) * scale;
  const float Vg = ((float)iy + 0.5f) * scale;

  v2f bop;                         // B matrix (4 x 16 pixels), wave-constant
  bop[0] = hi ? 1.0f : Ug;         // {Ug,Vg} in K0/K1 lanes, {1,0} in K2/K3 lanes
  bop[1] = hi ? 0.0f : Vg;
  const float qpix = Ug * Ug + Vg * Vg;
  v8f cinit;
#pragma unroll
  for (int j = 0; j < 8; ++j) cinit[j] = qpix;

  float zk[KN], d2k[KN];
  int ik[KN];
#pragma unroll
  for (int s = 0; s < KN; ++s) { zk[s] = BIGZ; d2k[s] = 0.0f; ik[s] = 0; }
  const float r2 = RADN * RADN;

  for (int c = 0; c < PN / 16; ++c) {
    // unconditional b128 load; component select via cndmask (no exec branches)
    float4 ra = sRec[c * 16 + l16];          // {-2U, -2V, q, z} for point c*16+l16
    v2f aop;
    aop[0] = hi ? ra.z : ra.x;               // K0=-2U / K2=q
    aop[1] = hi ? 0.0f : ra.y;               // K1=-2V / K3=0
    // D[m][n] = q_m - 2U_m*Ug_n - 2V_m*Vg_n + (Ug_n^2+Vg_n^2) = dist2
    v8f d = __builtin_amdgcn_wmma_f32_16x16x4_f32(
        false, aop, false, bop, (short)0, cinit, false, false);

    const int ibase = c * 16 + (hi ? 8 : 0); // my 8 point indices start here
    const int zrow = c * 4 + (hi ? 2 : 0);
    float4 za = sZ4[zrow];
    float4 zb = sZ4[zrow + 1];
    float zs[8] = {za.x, za.y, za.z, za.w, zb.x, zb.y, zb.z, zb.w};
#pragma unroll
    for (int j = 0; j < 8; ++j) {
      float dd = d[j];
      float zz = zs[j];
      if (dd < r2 && zz > 1e-3f && zz < zk[KN - 1]) {
        INSERT_K(zz, dd, ibase + j);
      }
    }
  }

  // snapshot partner half's sorted list, then merge (only lanes 0-15 finish)
  float pz[KN], pd[KN];
  int pi[KN];
#pragma unroll
  for (int t = 0; t < KN; ++t) {
    pz[t] = __shfl_xor(zk[t], 16, 32);
    pd[t] = __shfl_xor(d2k[t], 16, 32);
    pi[t] = __shfl_xor(ik[t], 16, 32);
  }

  if (!hi) {
#pragma unroll
    for (int t = 0; t < KN; ++t) {
      if (pz[t] < zk[KN - 1]) { INSERT_K(pz[t], pd[t], pi[t]); }
    }
    // front-to-back alpha compositing: w_k = a_k * prod_{j<k}(1-a_j)
    const float inv_r2 = 1.0f / r2;
    float T = 1.0f, cr = 0.0f, cg = 0.0f, cb = 0.0f;
#pragma unroll
    for (int k = 0; k < KN; ++k) {
      bool valid = zk[k] < BIGZ * 0.5f;
      float a = valid ? fminf(fmaxf(1.0f - d2k[k] * inv_r2, 0.0f), 1.0f) : 0.0f;
      float w = a * T;
      T *= (1.0f - a);
      if (valid && w > 0.0f) {
        const float* fp = feat + ((size_t)b * PN + ik[k]) * 3;
        cr += w * fp[0];
        cg += w * fp[1];
        cb += w * fp[2];
      }
    }
    float* img = out + ((size_t)b * IMGW * IMGW + pix) * 3;
    img[0] = cr; img[1] = cg; img[2] = cb;
    float* dep = out + (size_t)BN * IMGW * IMGW * 3;
    dep[(size_t)b * IMGW * IMGW + pix] = (zk[0] < BIGZ * 0.5f) ? zk[0] : -1.0f;
  }
}

// ---------------------------------------------------------------------------
extern "C" void kernel_launch(void* const* d_in, const int* in_sizes, int n_in,
                              void* d_out, int out_size, void* d_ws, size_t ws_size,
                              hipStream_t stream) {
  const float* points = (const float*)d_in[0];   // [B,P,3]
  const float* feats  = (const float*)d_in[1];   // [B,P,3]
  const float* camR   = (const float*)d_in[2];   // [B,3,3]
  const float* camT   = (const float*)d_in[3];   // [B,3]
  const float* fptr   = (const float*)d_in[4];   // scalar
  (void)in_sizes; (void)n_in; (void)out_size; (void)ws_size;

  float4* rec  = (float4*)d_ws;                                   // 64 KB
  float*  zarr = (float*)((char*)d_ws + (size_t)BN * PN * sizeof(float4)); // 16 KB

  pcr_project<<<(BN * PN + 255) / 256, 256, 0, stream>>>(points, camR, camT,
                                                         fptr, rec, zarr);
  dim3 grid(IMGW * IMGW / 128, BN);   // 128 pixels per 256-thread block
  pcr_render<<<grid, 256, 0, stream>>>(rec, zarr, feats, (float*)d_out);
}